// ST_LSTM_78486232367526
// MI455X (gfx1250) — compile-verified
//
#include <hip/hip_runtime.h>
#include <math.h>

// ---- problem constants (from reference) ----
#define BB   256
#define TIN  49
#define TOUT 25
#define NBONE 24
#define HH   128
#define FH   (5 * HH)          // 640 packed gate columns
#define MT   32                // batch rows per workgroup
#define NT   16                // hidden columns per workgroup
#define NTHREADS 320           // 10 wave32 waves: 2 m-subtiles x 5 gates

typedef __attribute__((ext_vector_type(2))) float v2f;
typedef __attribute__((ext_vector_type(4))) float f4;
typedef __attribute__((ext_vector_type(8))) float v8f;

__device__ __forceinline__ float sigm(float x) { return 1.0f / (1.0f + __expf(-x)); }

// ------------------------------------------------------------------
// Kernel 1: temporal mean reductions over TIN frames.
//   row_h0[nb][b][h] = mean_t hidden,  row_c = mean_t cell,
//   row_h1 = (global_t_state + sum_t hidden) / (TIN+1)
// ------------------------------------------------------------------
__global__ void k_time_reduce(const float* __restrict__ hs, const float* __restrict__ cs,
                              const float* __restrict__ gts,
                              float* __restrict__ row_h0, float* __restrict__ row_h1,
                              float* __restrict__ row_c) {
    int i = blockIdx.x * blockDim.x + threadIdx.x;     // over B*NB*H
    if (i >= BB * NBONE * HH) return;
    int h  = i % HH;
    int nb = (i / HH) % NBONE;
    int b  = i / (HH * NBONE);
    const float* ph = hs + ((size_t)b * TIN * NBONE + nb) * HH + h;
    const float* pc = cs + ((size_t)b * TIN * NBONE + nb) * HH + h;
    float sh = 0.f, sc = 0.f;
    for (int t = 0; t < TIN; ++t) {
        sh += ph[(size_t)t * NBONE * HH];
        sc += pc[(size_t)t * NBONE * HH];
    }
    float g = gts[((size_t)b * NBONE + nb) * HH + h];
    size_t o = ((size_t)nb * BB + b) * HH + h;
    row_h0[o] = sh * (1.0f / TIN);
    row_c [o] = sc * (1.0f / TIN);
    row_h1[o] = (sh + g) * (1.0f / (TIN + 1));
}

// ------------------------------------------------------------------
// Kernel 2: bone mean -> boundary column states col_h / col_c  [B][H]
// ------------------------------------------------------------------
__global__ void k_col_reduce(const float* __restrict__ row_h0, const float* __restrict__ row_c,
                             float* __restrict__ col_h, float* __restrict__ col_c) {
    int i = blockIdx.x * blockDim.x + threadIdx.x;     // over B*H
    if (i >= BB * HH) return;
    float sh = 0.f, sc = 0.f;
    for (int nb = 0; nb < NBONE; ++nb) {
        sh += row_h0[(size_t)nb * BB * HH + i];
        sc += row_c [(size_t)nb * BB * HH + i];
    }
    col_h[i] = sh * (1.0f / NBONE);
    col_c[i] = sc * (1.0f / NBONE);
}

// ------------------------------------------------------------------
// Kernel 3: one wavefront diagonal of ST-LSTM cells.
//   blockIdx.z = slot: [0,24) layer0 bone nb, [24,48) layer1 bone nb-24
//   layer0 cell (t,nb) at d = t+nb ; layer1 cell at d = t+nb+1
//   State grids H0/C0/H1/C1: [2 parity][NB][B][H]; write parity d&1,
//   read parity (d-1)&1 — race-free double buffering per diagonal.
// ------------------------------------------------------------------
__global__ __launch_bounds__(NTHREADS) void k_cell(
    const float* __restrict__ p,
    const float* __restrict__ U, const float* __restrict__ Wt, const float* __restrict__ Ws,
    const float* __restrict__ bias,
    const float* __restrict__ row_h0, const float* __restrict__ row_h1,
    const float* __restrict__ row_c,
    const float* __restrict__ col_h, const float* __restrict__ col_c,
    float* __restrict__ H0g, float* __restrict__ C0g,
    float* __restrict__ H1g, float* __restrict__ C1g,
    float* __restrict__ out, int d)
{
    __shared__ float A_lds[3][MT][HH];          // x, h_t, h_s tiles (48 KB)
    __shared__ float z_lds[2][5][16][16];       // gate tiles (10 KB)

    const int slot  = blockIdx.z;
    const int layer = (slot >= NBONE) ? 1 : 0;
    const int nb    = layer ? (slot - NBONE) : slot;
    const int t     = d - layer - nb;
    if (t < 0 || t >= TOUT) return;             // uniform early-exit

    const int b0  = blockIdx.y * MT;            // batch block
    const int n0  = blockIdx.x * NT;            // hidden-column slice
    const size_t GRID = (size_t)NBONE * BB * HH;
    const size_t par  = (size_t)(d & 1) * GRID;
    const size_t parp = (size_t)((d - 1) & 1) * GRID;

    // -------- resolve A-matrix sources (uniform per block) --------
    const float *xb, *htb, *hsb, *ctb, *csb;
    size_t xstride;
    if (layer == 0) {
        xb = p + ((size_t)b0 * TOUT + t) * NBONE * HH + (size_t)nb * HH;
        xstride = (size_t)TOUT * NBONE * HH;
        if (t == 0) { htb = row_h0 + ((size_t)nb * BB + b0) * HH;
                      ctb = row_c  + ((size_t)nb * BB + b0) * HH; }
        else        { htb = H0g + parp + ((size_t)nb * BB + b0) * HH;
                      ctb = C0g + parp + ((size_t)nb * BB + b0) * HH; }
        if (nb == 0) { hsb = col_h + (size_t)b0 * HH;
                       csb = col_c + (size_t)b0 * HH; }
        else         { hsb = H0g + parp + ((size_t)(nb - 1) * BB + b0) * HH;
                       csb = C0g + parp + ((size_t)(nb - 1) * BB + b0) * HH; }
    } else {
        xb = H0g + parp + ((size_t)nb * BB + b0) * HH;   // layer-0 output at (t,nb)
        xstride = HH;
        if (t == 0) { htb = row_h1 + ((size_t)nb * BB + b0) * HH;
                      ctb = row_c  + ((size_t)nb * BB + b0) * HH; }
        else        { htb = H1g + parp + ((size_t)nb * BB + b0) * HH;
                      ctb = C1g + parp + ((size_t)nb * BB + b0) * HH; }
        if (nb == 0) { hsb = col_h + (size_t)b0 * HH;
                       csb = col_c + (size_t)b0 * HH; }
        else         { hsb = H1g + parp + ((size_t)(nb - 1) * BB + b0) * HH;
                       csb = C1g + parp + ((size_t)(nb - 1) * BB + b0) * HH; }
    }

    const int tid = threadIdx.x;

    // -------- stage A matrices into LDS (float4, coalesced) --------
    for (int i = tid; i < MT * (HH / 4); i += NTHREADS) {
        int r = i / (HH / 4), kv = i % (HH / 4);
        *(f4*)&A_lds[0][r][kv * 4] = *((const f4*)(xb + (size_t)r * xstride) + kv);
    }
    for (int i = tid; i < MT * (HH / 4); i += NTHREADS) {
        int r = i / (HH / 4), kv = i % (HH / 4);
        *(f4*)&A_lds[1][r][kv * 4] = *((const f4*)(htb + (size_t)r * HH) + kv);
    }
    for (int i = tid; i < MT * (HH / 4); i += NTHREADS) {
        int r = i / (HH / 4), kv = i % (HH / 4);
        *(f4*)&A_lds[2][r][kv * 4] = *((const f4*)(hsb + (size_t)r * HH) + kv);
    }
    __syncthreads();

    // -------- per-wave WMMA GEMM: one 16x16 gate tile per wave --------
    const int lane  = tid & 31;
    const int wave  = tid >> 5;
    const int msub  = wave & 1;          // which 16-row batch subtile
    const int gate  = wave >> 1;         // 0:i 1:f_s 2:f_t 3:o 4:g
    const int nloc  = lane & 15;
    const int khalf = (lane >> 4) * 2;   // K-pair select for 16x16x4 A/B frags
    const int col   = gate * HH + n0 + nloc;        // packed 5H weight column
    const int arow  = msub * 16 + nloc;             // A row = M index

    const size_t wbase = ((size_t)layer * NBONE + nb) * HH * FH;
    const float* Wsrc[3] = { U + wbase + col, Wt + wbase + col, Ws + wbase + col };

    const float bv = bias[((size_t)layer * NBONE + nb) * FH + col];
    v8f acc = { bv, bv, bv, bv, bv, bv, bv, bv };   // bias broadcast over rows

    #pragma unroll
    for (int s = 0; s < 3; ++s) {
        const float* W    = Wsrc[s];
        const float* Arow = &A_lds[s][arow][0];
        #pragma unroll
        for (int kk = 0; kk < HH; kk += 4) {
            int k0 = kk + khalf;
            v2f a = *(const v2f*)&Arow[k0];          // A[M][k0], A[M][k0+1]
            v2f bf;
            bf.x = W[(size_t)k0 * FH];               // B[k0][n]
            bf.y = W[(size_t)(k0 + 1) * FH];         // B[k0+1][n]
            acc = __builtin_amdgcn_wmma_f32_16x16x4_f32(
                      false, a, false, bf, (short)0, acc, false, false);
        }
    }

    // spill gate tile to LDS: VGPR v -> row v + (lane>>4)*8, col = lane&15
    #pragma unroll
    for (int v = 0; v < 8; ++v)
        z_lds[msub][gate][v + (lane >> 4) * 8][nloc] = acc[v];
    __syncthreads();

    // -------- gates -> (h, c) and state/out writes --------
    for (int i = tid; i < MT * NT; i += NTHREADS) {
        int r = i / NT, j = i % NT;
        int rs = r >> 4, rl = r & 15;
        float zi  = z_lds[rs][0][rl][j];
        float zfs = z_lds[rs][1][rl][j];
        float zft = z_lds[rs][2][rl][j];
        float zo  = z_lds[rs][3][rl][j];
        float zg  = z_lds[rs][4][rl][j];
        float ctv = ctb[(size_t)r * HH + n0 + j];
        float csv = csb[(size_t)r * HH + n0 + j];
        float c = sigm(zi) * tanhf(zg) + sigm(zft) * ctv + sigm(zfs) * csv;
        float h = sigm(zo) * tanhf(c);
        int b = b0 + r;
        size_t so = par + ((size_t)nb * BB + b) * HH + n0 + j;
        if (layer == 0) {
            H0g[so] = h; C0g[so] = c;
        } else {
            H1g[so] = h; C1g[so] = c;
            size_t oo = (((size_t)b * TOUT + t) * NBONE + nb) * HH + n0 + j;
            out[oo] = h;
            out[(size_t)BB * TOUT * NBONE * HH + oo] = c;
        }
    }
}

// ------------------------------------------------------------------
extern "C" void kernel_launch(void* const* d_in, const int* in_sizes, int n_in,
                              void* d_out, int out_size, void* d_ws, size_t ws_size,
                              hipStream_t stream) {
    const float* hs   = (const float*)d_in[0];   // hidden_states [B,TIN,NB,H]
    const float* cs   = (const float*)d_in[1];   // cell_states
    const float* gts  = (const float*)d_in[2];   // global_t_state [B,NB,H]
    const float* p    = (const float*)d_in[3];   // p [B,TOUT,NB,H]
    const float* U    = (const float*)d_in[4];   // [2,NB,H,5H]
    const float* Wt   = (const float*)d_in[5];
    const float* Ws   = (const float*)d_in[6];
    const float* bias = (const float*)d_in[7];   // [2,NB,5H]
    float* out = (float*)d_out;

    // workspace carve-up (floats)
    const size_t ROW = (size_t)NBONE * BB * HH;     // 786432
    const size_t COL = (size_t)BB * HH;             // 32768
    const size_t GRID2 = 2 * ROW;                   // double-buffered state slab
    float* ws      = (float*)d_ws;
    float* row_h0  = ws;
    float* row_h1  = row_h0 + ROW;
    float* row_c   = row_h1 + ROW;
    float* col_h   = row_c  + ROW;
    float* col_c   = col_h  + COL;
    float* H0g     = col_c  + COL;
    float* C0g     = H0g + GRID2;
    float* H1g     = C0g + GRID2;
    float* C1g     = H1g + GRID2;

    k_time_reduce<<<(BB * NBONE * HH + 255) / 256, 256, 0, stream>>>(
        hs, cs, gts, row_h0, row_h1, row_c);
    k_col_reduce<<<(BB * HH + 255) / 256, 256, 0, stream>>>(
        row_h0, row_c, col_h, col_c);

    // wavefront over anti-diagonals: d = t+nb (layer0), t+nb+1 (layer1)
    dim3 grid(HH / NT, BB / MT, 2 * NBONE);         // (8, 8, 48)
    for (int d = 0; d <= (TOUT - 1) + (NBONE - 1) + 1; ++d) {
        k_cell<<<grid, NTHREADS, 0, stream>>>(
            p, U, Wt, Ws, bias, row_h0, row_h1, row_c, col_h, col_c,
            H0g, C0g, H1g, C1g, out, d);
    }
}